// gaeEncoder_69475390980569
// MI455X (gfx1250) — compile-verified
//
#include <hip/hip_runtime.h>

typedef float v2f __attribute__((ext_vector_type(2)));
typedef float v8f __attribute__((ext_vector_type(8)));

#define NNODES 200000

// ---------------- degree / inv-degree ----------------

__global__ void deg_kernel(const int* __restrict__ dst, unsigned int* __restrict__ deg, int E) {
    int e = blockIdx.x * blockDim.x + threadIdx.x;
    if (e < E) {
        __hip_atomic_fetch_add(&deg[dst[e]], 1u, __ATOMIC_RELAXED,
                               __HIP_MEMORY_SCOPE_AGENT);
    }
}

__global__ void invdeg_kernel(const unsigned int* __restrict__ deg, float* __restrict__ inv,
                              int n) {
    int i = blockIdx.x * blockDim.x + threadIdx.x;
    if (i < n) {
        unsigned int d = deg[i];
        inv[i] = (d > 0u) ? (1.0f / (float)d) : 0.0f;
    }
}

// ---------------- scatter-add: agg[dst] += x[src] ----------------
// one thread per (edge, 4-feature group): one b128 gather + 4 fp32 atomics.
// D = 1<<logD; groups per edge = 1<<(logD-2).

__global__ void scatter_kernel(const float* __restrict__ x, const int* __restrict__ src,
                               const int* __restrict__ dst, float* __restrict__ agg,
                               int E, int logD) {
    int logG = logD - 2;                       // log2(D/4)
    long long idx = (long long)blockIdx.x * blockDim.x + threadIdx.x;
    long long total = (long long)E << logG;
    if (idx >= total) return;
    int e = (int)(idx >> logG);
    int c = (int)(idx & ((1 << logG) - 1)) << 2;   // feature offset, multiple of 4
    int D = 1 << logD;

    long long srow = (long long)src[e] * D + c;
    long long drow = (long long)dst[e] * D + c;

    float4 v = *reinterpret_cast<const float4*>(x + srow);   // global_load_b128
    float* a = agg + drow;
    __hip_atomic_fetch_add(a + 0, v.x, __ATOMIC_RELAXED, __HIP_MEMORY_SCOPE_AGENT);
    __hip_atomic_fetch_add(a + 1, v.y, __ATOMIC_RELAXED, __HIP_MEMORY_SCOPE_AGENT);
    __hip_atomic_fetch_add(a + 2, v.z, __ATOMIC_RELAXED, __HIP_MEMORY_SCOPE_AGENT);
    __hip_atomic_fetch_add(a + 3, v.w, __ATOMIC_RELAXED, __HIP_MEMORY_SCOPE_AGENT);
}

// ---------------- fused SAGE GEMM via V_WMMA_F32_16X16X4_F32 ----------------
// out[m, :] = act( (inv[m]*agg[m,:]) @ Wl + bl + x[m,:] @ Wr )
// One wave computes a 16-row stripe across NTILES 16-col tiles.
// Weights staged in LDS, zero-padded to Npad = NTILES*16 columns.

template <int NTILES>
__global__ void sage_gemm_kernel(const float* __restrict__ agg, const float* __restrict__ xin,
                                 const float* __restrict__ invdeg,
                                 const float* __restrict__ Wl, const float* __restrict__ bl,
                                 const float* __restrict__ Wr, float* __restrict__ out,
                                 int K, int Nout, int Mtiles, int do_relu) {
    constexpr int Npad = NTILES * 16;
    extern __shared__ float lds[];
    float* ldsWl = lds;
    float* ldsWr = lds + K * Npad;

    // cooperative zero-padded weight stage (K*Npad each)
    int tid = threadIdx.x;
    for (int i = tid; i < K * Npad; i += blockDim.x) {
        int k = i / Npad;
        int n = i % Npad;
        float wl = (n < Nout) ? Wl[k * Nout + n] : 0.0f;
        float wr = (n < Nout) ? Wr[k * Nout + n] : 0.0f;
        ldsWl[i] = wl;
        ldsWr[i] = wr;
    }
    __syncthreads();

    int wave = tid >> 5;
    int lane = tid & 31;
    int mtile = blockIdx.x * (blockDim.x >> 5) + wave;
    if (mtile >= Mtiles) return;  // uniform per wave: EXEC stays all-ones for WMMA

    int row16 = lane & 15;   // A: row within tile; B/C: column within tile
    int khalf = lane >> 4;   // A/B: which K pair; C: row offset +8
    int m = mtile * 16 + row16;
    float sc = invdeg[m];

    const float* aggRow = agg + (long long)m * K;
    const float* xRow   = xin + (long long)m * K;

    v8f acc[NTILES];
#pragma unroll
    for (int t = 0; t < NTILES; ++t) {
        v8f z = {0.f, 0.f, 0.f, 0.f, 0.f, 0.f, 0.f, 0.f};
        acc[t] = z;
    }

    for (int k0 = 0; k0 < K; k0 += 4) {
        int ka = k0 + 2 * khalf;          // this lane's K pair: ka, ka+1
        v2f aAgg, aX;
        aAgg.x = aggRow[ka] * sc;
        aAgg.y = aggRow[ka + 1] * sc;
        aX.x = xRow[ka];
        aX.y = xRow[ka + 1];
#pragma unroll
        for (int t = 0; t < NTILES; ++t) {
            int col = t * 16 + row16;
            v2f bL, bR;
            bL.x = ldsWl[ka * Npad + col];
            bL.y = ldsWl[(ka + 1) * Npad + col];
            bR.x = ldsWr[ka * Npad + col];
            bR.y = ldsWr[(ka + 1) * Npad + col];
            acc[t] = __builtin_amdgcn_wmma_f32_16x16x4_f32(
                false, aAgg, false, bL, (short)0, acc[t], false, false);
            acc[t] = __builtin_amdgcn_wmma_f32_16x16x4_f32(
                false, aX, false, bR, (short)0, acc[t], false, false);
        }
    }

    // C/D layout: VGPR v -> row (v + 8*khalf), col = row16 (+ tile offset)
#pragma unroll
    for (int t = 0; t < NTILES; ++t) {
        int col = t * 16 + row16;
        if (col < Nout) {
            float bias = bl[col];
#pragma unroll
            for (int v = 0; v < 8; ++v) {
                int row = mtile * 16 + v + 8 * khalf;
                float val = acc[t][v] + bias;
                if (do_relu) val = fmaxf(val, 0.0f);
                out[(long long)row * Nout + col] = val;
            }
        }
    }
}

// ---------------- host-side launch ----------------

extern "C" void kernel_launch(void* const* d_in, const int* in_sizes, int n_in,
                              void* d_out, int out_size, void* d_ws, size_t ws_size,
                              hipStream_t stream) {
    const float* x   = (const float*)d_in[0];
    const int*  eidx = (const int*)d_in[1];     // [2, E] int32
    int E = in_sizes[1] / 2;
    const int* src = eidx;
    const int* dst = eidx + E;

    const float* Wl1 = (const float*)d_in[2];
    const float* bl1 = (const float*)d_in[3];
    const float* Wr1 = (const float*)d_in[4];
    const float* Wl2 = (const float*)d_in[5];
    const float* bl2 = (const float*)d_in[6];
    const float* Wr2 = (const float*)d_in[7];
    const float* Wl3 = (const float*)d_in[8];
    const float* bl3 = (const float*)d_in[9];
    const float* Wr3 = (const float*)d_in[10];

    // workspace layout
    float* ws    = (float*)d_ws;
    unsigned int* deg = (unsigned int*)ws;          // N (u32 counts)
    float* inv   = ws + NNODES;                     // N
    float* aggb  = inv + NNODES;                    // N*128 (reused each layer)
    float* h1    = aggb + (size_t)NNODES * 128;     // N*64
    float* h2    = h1 + (size_t)NNODES * 64;        // N*32

    const int TB = 256;
    int Mtiles = NNODES / 16;                       // 12500, exact
    int gemmBlocks = (Mtiles + 7) / 8;              // 8 waves / block

    // degrees
    hipMemsetAsync(deg, 0, NNODES * sizeof(unsigned int), stream);
    deg_kernel<<<(E + TB - 1) / TB, TB, 0, stream>>>(dst, deg, E);
    invdeg_kernel<<<(NNODES + TB - 1) / TB, TB, 0, stream>>>(deg, inv, NNODES);

    // ---- layer 1: K=128, N=64, relu ----
    hipMemsetAsync(aggb, 0, (size_t)NNODES * 128 * sizeof(float), stream);
    {
        long long total = (long long)E << 5;    // E * D/4, D=128
        scatter_kernel<<<(int)((total + TB - 1) / TB), TB, 0, stream>>>(x, src, dst, aggb, E, 7);
    }
    sage_gemm_kernel<4><<<gemmBlocks, TB, 128 * 64 * 2 * sizeof(float), stream>>>(
        aggb, x, inv, Wl1, bl1, Wr1, h1, 128, 64, Mtiles, 1);

    // ---- layer 2: K=64, N=32, relu ----
    hipMemsetAsync(aggb, 0, (size_t)NNODES * 64 * sizeof(float), stream);
    {
        long long total = (long long)E << 4;    // E * D/4, D=64
        scatter_kernel<<<(int)((total + TB - 1) / TB), TB, 0, stream>>>(h1, src, dst, aggb, E, 6);
    }
    sage_gemm_kernel<2><<<gemmBlocks, TB, 64 * 32 * 2 * sizeof(float), stream>>>(
        aggb, h1, inv, Wl2, bl2, Wr2, h2, 64, 32, Mtiles, 1);

    // ---- layer 3: K=32, N=20 (pad to 32), no relu ----
    hipMemsetAsync(aggb, 0, (size_t)NNODES * 32 * sizeof(float), stream);
    {
        long long total = (long long)E << 3;    // E * D/4, D=32
        scatter_kernel<<<(int)((total + TB - 1) / TB), TB, 0, stream>>>(h2, src, dst, aggb, E, 5);
    }
    sage_gemm_kernel<2><<<gemmBlocks, TB, 32 * 32 * 2 * sizeof(float), stream>>>(
        aggb, h2, inv, Wl3, bl3, Wr3, (float*)d_out, 32, 20, Mtiles, 0);
}